// MultiHeadAttention_6966436954266
// MI455X (gfx1250) — compile-verified
//
#include <hip/hip_runtime.h>
#include <math.h>

// Problem constants (match reference)
#define B_   2
#define N_   20000
#define D_   256
#define H_   8
#define DH_  32
#define DEG_ 8
#define EPS_ 1e-3f

// 260 floats: multiple of 4 (keeps 16B alignment for b128 LDS stores) and
// 260 % 64 == 4, so A-fragment reads map lane L -> bank pair {4L+k, 4L+k+1}:
// all 32 lanes of a wave hit 64 distinct banks for the ds_load b64 pairs.
#define LDSTRIDE (D_ + 4)

typedef __attribute__((ext_vector_type(2))) float v2f;
typedef __attribute__((ext_vector_type(8))) float v8f;

__device__ __forceinline__ float waveReduceSum(float v) {
    // wave32: full-wave butterfly reduction
    #pragma unroll
    for (int off = 16; off > 0; off >>= 1)
        v += __shfl_xor(v, off, 32);
    return v;
}

// Load 16 rows x 256 cols into LDS with b128 loads (4 float4 per thread).
__device__ __forceinline__ void loadTile16(const float* __restrict__ src,
                                           float* __restrict__ lds, int t) {
    const float4* s4 = (const float4*)src;
    #pragma unroll
    for (int i = t; i < 16 * (D_ / 4); i += 256) {
        int r  = i >> 6;        // / 64
        int c4 = i & 63;
        float4 val = s4[r * (D_ / 4) + c4];
        *(float4*)&lds[r * LDSTRIDE + c4 * 4] = val;
    }
}

// Per-row LayerNorm over a 16-row LDS tile; writes normalized values to dst.
__device__ __forceinline__ void layerNormTile16(const float* __restrict__ srcLds,
                                                float* __restrict__ dstLds,
                                                float* __restrict__ s_mu,
                                                float* __restrict__ s_rs,
                                                const float* __restrict__ gamma,
                                                const float* __restrict__ beta,
                                                int wave, int lane, int t) {
    #pragma unroll
    for (int rr = 0; rr < 2; ++rr) {
        int r = wave * 2 + rr;
        float s = 0.f;
        for (int c = lane; c < D_; c += 32) s += srcLds[r * LDSTRIDE + c];
        s = waveReduceSum(s);
        float m = s * (1.0f / D_);
        float vs = 0.f;
        for (int c = lane; c < D_; c += 32) {
            float d = srcLds[r * LDSTRIDE + c] - m; vs += d * d;
        }
        vs = waveReduceSum(vs);
        if (lane == 0) { s_mu[r] = m; s_rs[r] = rsqrtf(vs * (1.0f / D_) + EPS_); }
    }
    __syncthreads();
    {
        int r  = t >> 4;
        int c0 = (t & 15) * 16;
        float m = s_mu[r], sc = s_rs[r];
        #pragma unroll
        for (int c = c0; c < c0 + 16; ++c)
            dstLds[r * LDSTRIDE + c] = (srcLds[r * LDSTRIDE + c] - m) * sc * gamma[c] + beta[c];
    }
    __syncthreads();
}

// ---------------------------------------------------------------------------
// Kernel 1: fused LayerNorm1 + QKV projection GEMM (fp32 WMMA 16x16x4)
// grid = (B*N)/16 blocks, 256 threads (8 waves). Each block: 16 rows.
// Each wave computes column tiles {wave, wave+8} simultaneously so the A
// fragment is loaded once per two WMMAs.
// ---------------------------------------------------------------------------
__global__ __launch_bounds__(256)
void qkv_kernel(const float* __restrict__ x,
                const float* __restrict__ wq, const float* __restrict__ bq,
                const float* __restrict__ wk, const float* __restrict__ bk,
                const float* __restrict__ wv, const float* __restrict__ bv,
                const float* __restrict__ g1, const float* __restrict__ be1,
                float* __restrict__ q, float* __restrict__ k, float* __restrict__ v)
{
    __shared__ __align__(16) float lds[16 * LDSTRIDE];
    __shared__ float s_mu[16], s_rs[16];

    const int t    = threadIdx.x;
    const int wave = t >> 5;
    const int lane = t & 31;
    const int row0 = blockIdx.x * 16;

    // Warm L2 for the weight slices this wave will read (global_prefetch_b8).
    __builtin_prefetch(wq + wave * 16 + lane, 0, 1);
    __builtin_prefetch(wk + wave * 16 + lane, 0, 1);
    __builtin_prefetch(wv + wave * 16 + lane, 0, 1);

    loadTile16(x + (size_t)row0 * D_, lds, t);
    __syncthreads();
    layerNormTile16(lds, lds, s_mu, s_rs, g1, be1, wave, lane, t);

    const int half = lane >> 4;   // 0: lanes 0-15, 1: lanes 16-31
    const int lrow = lane & 15;
    const int n0   = wave * 16;   // first column tile of this wave
    const int n1   = n0 + 128;    // second column tile (wave+8)

    #pragma unroll 1
    for (int m = 0; m < 3; ++m) {
        const float* __restrict__ W  = (m == 0) ? wq : (m == 1) ? wk : wv;
        const float* __restrict__ Bb = (m == 0) ? bq : (m == 1) ? bk : bv;
        float* __restrict__ O        = (m == 0) ? q  : (m == 1) ? k  : v;

        v8f acc0 = {}, acc1 = {};
        #pragma unroll 4
        for (int k0 = 0; k0 < D_; k0 += 4) {
            // A 16x4 fp32 fragment: lanes 0-15 K={k0,k0+1}, lanes 16-31 K={k0+2,k0+3}
            v2f a;
            a.x = lds[lrow * LDSTRIDE + k0 + 2 * half + 0];
            a.y = lds[lrow * LDSTRIDE + k0 + 2 * half + 1];
            // B 4x16 fp32 fragments for both column tiles (row-striped, mirrors A)
            const float* __restrict__ Wp = W + (size_t)(k0 + 2 * half) * D_;
            v2f b0, b1;
            b0.x = Wp[n0 + lrow];
            b0.y = Wp[D_ + n0 + lrow];
            b1.x = Wp[n1 + lrow];
            b1.y = Wp[D_ + n1 + lrow];
            acc0 = __builtin_amdgcn_wmma_f32_16x16x4_f32(
                false, a, false, b0, (short)0, acc0, false, false);
            acc1 = __builtin_amdgcn_wmma_f32_16x16x4_f32(
                false, a, false, b1, (short)0, acc1, false, false);
        }

        const float bias0 = Bb[n0 + lrow];
        const float bias1 = Bb[n1 + lrow];
        #pragma unroll
        for (int r = 0; r < 8; ++r) {
            // C layout: VGPR r -> row r (lanes 0-15) / row r+8 (lanes 16-31); col = lane%16
            size_t orow = (size_t)(row0 + r + 8 * half);
            O[orow * D_ + n0 + lrow] = acc0[r] + bias0;
            O[orow * D_ + n1 + lrow] = acc1[r] + bias1;
        }
    }
}

// ---------------------------------------------------------------------------
// Kernel 2: edge attention + residual. grid = B*N blocks, 256 threads.
// Edges with src==n are contiguous [n*DEG, n*DEG+DEG) -> no atomics.
// wave w == head w (dh=32 == wave32: one head per wave exactly).
// ---------------------------------------------------------------------------
__global__ __launch_bounds__(256)
void attn_kernel(const float* __restrict__ x,
                 const int* __restrict__ edges,         // [E,2] int32
                 const float* __restrict__ g1, const float* __restrict__ be1,
                 const float* __restrict__ q, const float* __restrict__ k,
                 const float* __restrict__ v,
                 float* __restrict__ concat)
{
    __shared__ float red[8];

    const int t    = threadIdx.x;
    const int wave = t >> 5;
    const int lane = t & 31;
    const int bid  = blockIdx.x;
    const int b    = bid / N_;
    const int n    = bid - b * N_;
    const size_t rowOff = (size_t)bid * D_;   // (b*N + n)*D

    // --- recompute xn for this row (LN1) ---
    float xv = x[rowOff + t];
    float s = waveReduceSum(xv);
    if (lane == 0) red[wave] = s;
    __syncthreads();
    float mean = 0.f;
    #pragma unroll
    for (int i = 0; i < 8; ++i) mean += red[i];
    mean *= (1.0f / D_);
    __syncthreads();
    float dlt = xv - mean;
    float s2 = waveReduceSum(dlt * dlt);
    if (lane == 0) red[wave] = s2;
    __syncthreads();
    float var = 0.f;
    #pragma unroll
    for (int i = 0; i < 8; ++i) var += red[i];
    var *= (1.0f / D_);
    const float xn = dlt * rsqrtf(var + EPS_) * g1[t] + be1[t];

    // --- attention over the 8 contiguous edges with src == n ---
    const float qv = q[rowOff + t];
    float num = 0.f, den = 0.f;
    const float scale = 0.1767766952966369f;   // 1/sqrt(32)
    #pragma unroll
    for (int j = 0; j < DEG_; ++j) {
        const int e   = n * DEG_ + j;          // uniform across block -> scalar load
        const int dst = edges[2 * e + 1];
        const size_t drow = ((size_t)b * N_ + dst) * D_;
        const float kv = k[drow + t];
        const float dot = waveReduceSum(qv * kv);   // per-head dot product
        const float w   = expf(dot * scale);
        num += v[drow + t] * w;
        den += w;
    }
    concat[rowOff + t] = xn + num / den;
}

// ---------------------------------------------------------------------------
// Kernel 3: fused LayerNorm2 + output projection GEMM + ReLU + residual.
// grid = (B*N)/16 blocks, 256 threads. Same dual-tile WMMA scheme.
// ---------------------------------------------------------------------------
__global__ __launch_bounds__(256)
void out_kernel(const float* __restrict__ concat,
                const float* __restrict__ wo, const float* __restrict__ bo,
                const float* __restrict__ g2, const float* __restrict__ be2,
                float* __restrict__ out)
{
    __shared__ __align__(16) float raw[16 * LDSTRIDE];   // concat (residual)
    __shared__ __align__(16) float nrm[16 * LDSTRIDE];   // LN2(concat) = GEMM A
    __shared__ float s_mu[16], s_rs[16];

    const int t    = threadIdx.x;
    const int wave = t >> 5;
    const int lane = t & 31;
    const int row0 = blockIdx.x * 16;

    __builtin_prefetch(wo + wave * 16 + lane, 0, 1);

    loadTile16(concat + (size_t)row0 * D_, raw, t);
    __syncthreads();
    layerNormTile16(raw, nrm, s_mu, s_rs, g2, be2, wave, lane, t);

    const int half = lane >> 4;
    const int lrow = lane & 15;
    const int n0   = wave * 16;
    const int n1   = n0 + 128;

    v8f acc0 = {}, acc1 = {};
    #pragma unroll 4
    for (int k0 = 0; k0 < D_; k0 += 4) {
        v2f a;
        a.x = nrm[lrow * LDSTRIDE + k0 + 2 * half + 0];
        a.y = nrm[lrow * LDSTRIDE + k0 + 2 * half + 1];
        const float* __restrict__ Wp = wo + (size_t)(k0 + 2 * half) * D_;
        v2f b0, b1;
        b0.x = Wp[n0 + lrow];
        b0.y = Wp[D_ + n0 + lrow];
        b1.x = Wp[n1 + lrow];
        b1.y = Wp[D_ + n1 + lrow];
        acc0 = __builtin_amdgcn_wmma_f32_16x16x4_f32(
            false, a, false, b0, (short)0, acc0, false, false);
        acc1 = __builtin_amdgcn_wmma_f32_16x16x4_f32(
            false, a, false, b1, (short)0, acc1, false, false);
    }

    const float bias0 = bo[n0 + lrow];
    const float bias1 = bo[n1 + lrow];
    #pragma unroll
    for (int r = 0; r < 8; ++r) {
        int rlocal = r + 8 * half;
        size_t orow = (size_t)(row0 + rlocal);
        float v0 = fmaxf(acc0[r] + bias0, 0.0f) + raw[rlocal * LDSTRIDE + n0 + lrow];
        float v1 = fmaxf(acc1[r] + bias1, 0.0f) + raw[rlocal * LDSTRIDE + n1 + lrow];
        out[orow * D_ + n0 + lrow] = v0;
        out[orow * D_ + n1 + lrow] = v1;
    }
}

// ---------------------------------------------------------------------------
extern "C" void kernel_launch(void* const* d_in, const int* in_sizes, int n_in,
                              void* d_out, int out_size, void* d_ws, size_t ws_size,
                              hipStream_t stream) {
    (void)in_sizes; (void)n_in; (void)out_size; (void)ws_size;

    const float* x      = (const float*)d_in[0];
    const int*   edges  = (const int*)  d_in[1];
    const float* wq     = (const float*)d_in[2];
    const float* bq     = (const float*)d_in[3];
    const float* wk     = (const float*)d_in[4];
    const float* bk     = (const float*)d_in[5];
    const float* wv     = (const float*)d_in[6];
    const float* bv     = (const float*)d_in[7];
    const float* wo     = (const float*)d_in[8];
    const float* bo     = (const float*)d_in[9];
    const float* gamma1 = (const float*)d_in[10];
    const float* beta1  = (const float*)d_in[11];
    const float* gamma2 = (const float*)d_in[12];
    const float* beta2  = (const float*)d_in[13];

    float* outp = (float*)d_out;

    const size_t SZ = (size_t)B_ * N_ * D_;   // 10.24M floats per tensor
    float* q      = (float*)d_ws;
    float* k      = q + SZ;
    float* v      = k + SZ;
    float* concat = v + SZ;

    const int rowBlocks = (B_ * N_) / 16;     // 2500

    qkv_kernel<<<rowBlocks, 256, 0, stream>>>(x, wq, bq, wk, bk, wv, bv,
                                              gamma1, beta1, q, k, v);
    attn_kernel<<<B_ * N_, 256, 0, stream>>>(x, edges, gamma1, beta1,
                                             q, k, v, concat);
    out_kernel<<<rowBlocks, 256, 0, stream>>>(concat, wo, bo, gamma2, beta2, outp);
}